// MetricNet_42838003810957
// MI455X (gfx1250) — compile-verified
//
#include <hip/hip_runtime.h>
#include <hip/hip_bf16.h>
#include <math.h>

// MetricNet layer scan for MI455X (gfx1250, wave32).
//
// Sequential nonlinear recurrence -> no matmul structure -> WMMA is not
// applicable. This is a VALU-f32 throughput problem. CDNA5-specific paths
// used: global_load_async_to_lds_b32 (ASYNCcnt) to stage B[] into LDS,
// s_wait_asynccnt, wave32-sized blocks, LDS-broadcast per-layer constants.
//
// All divisions are hoisted out of the 512-deep inner loop:
//   per-layer (LDS, once per block): A, C0, inv1, D0, E
//   per-element (once):              io = 1/omega, wo = dz*omega, w2 = 2*wo
// Inner loop: ~10 FMAs / element / layer, zero divisions.

#define NL     512
#define DZ     (0.9f / 512.0f)
#define MU2    1.0f
#define TPB    256

__global__ __launch_bounds__(TPB) void metricnet_scan(
    const float* __restrict__ Re_s, const float* __restrict__ Im_s,
    const float* __restrict__ omega, const float* __restrict__ PiTp,
    const float* __restrict__ B, float* __restrict__ out, int n)
{
    __shared__ float  sB[NL + 4];   // raw B[0..512]
    __shared__ float4 sC[NL];       // (A, C0, inv1, D0) per layer
    __shared__ float  sE[NL];       // E per layer

    const int tid = threadIdx.x;

    // ---- Stage 1: async-copy B[0..512] global -> LDS (CDNA5 ASYNCcnt path)
    {
        const float* g0 = B + tid;
        const float* g1 = B + tid + TPB;
        unsigned l0 = (unsigned)(unsigned long long)(const void*)&sB[tid];
        unsigned l1 = (unsigned)(unsigned long long)(const void*)&sB[tid + TPB];
        asm volatile("global_load_async_to_lds_b32 %0, %1, off"
                     :: "v"(l0), "v"(g0) : "memory");
        asm volatile("global_load_async_to_lds_b32 %0, %1, off"
                     :: "v"(l1), "v"(g1) : "memory");
        if (tid == 0) {
            const float* g2 = B + NL;
            unsigned l2 = (unsigned)(unsigned long long)(const void*)&sB[NL];
            asm volatile("global_load_async_to_lds_b32 %0, %1, off"
                         :: "v"(l2), "v"(g2) : "memory");
        }
        asm volatile("s_wait_asynccnt 0x0" ::: "memory");
    }
    __syncthreads();

    // ---- Stage 2: per-layer batch-invariant constants (all divisions here)
    const float PiT = PiTp[0];
    for (int i = tid; i < NL; i += TPB) {
        float b1   = sB[i];
        float b2   = sB[i + 1];
        float z    = DZ * (float)i;            // Z_INI = 0
        float omz  = 1.0f - z;
        float inv1 = 1.0f / (PiT * omz);
        float inv2 = inv1 / omz;
        float g    = 1.0f - b2 / b1;
        float A    = 1.0f + g;
        float C0   = g * inv1 - DZ * inv2;
        float D0   = 1.0f / (b1 * b1) - inv2 / PiT;
        float E    = DZ * ((z * z * MU2) / b1);
        sC[i] = make_float4(A, C0, inv1, D0);
        sE[i] = E;
    }
    __syncthreads();

    // ---- Stage 3: the scan, 2 independent elements per thread (ILP)
    const int base = (int)blockIdx.x * (2 * TPB);
    const int e0 = base + tid;
    const int e1 = base + tid + TPB;
    const bool v0 = (e0 < n);
    const bool v1 = (e1 < n);

    float Re0 = 0.f, Im0 = 0.f, om0 = 1.f;
    float Re1 = 0.f, Im1 = 0.f, om1 = 1.f;
    if (v0) { Re0 = Re_s[e0]; Im0 = Im_s[e0]; om0 = omega[e0]; }
    if (v1) { Re1 = Re_s[e1]; Im1 = Im_s[e1]; om1 = omega[e1]; }

    const float wo0 = DZ * om0, w20 = 2.0f * wo0, io0 = 1.0f / om0;
    const float wo1 = DZ * om1, w21 = 2.0f * wo1, io1 = 1.0f / om1;

#pragma unroll 4
    for (int l = 0; l < NL; ++l) {
        const float4 c  = sC[l];        // ds_load_b128, broadcast
        const float  E  = sE[l];        // ds_load_b32,  broadcast
        const float  i2 = c.z + c.z;    // 2*inv1 (shared by both elements)

        // element 0
        {
            float t  = Re0 + c.z;
            float p  = fmaf(c.x, Re0, c.y);
            float rn = fmaf(w20 * Im0, t, p);
            float u  = fmaf(Im0, Im0, c.w);
            u        = fmaf(-rn, rn + i2, u);
            Im0      = fmaf(wo0, u, fmaf(c.x, Im0, -(E * io0)));
            Re0      = rn;
        }
        // element 1 (independent chain -> hides FMA latency)
        {
            float t  = Re1 + c.z;
            float p  = fmaf(c.x, Re1, c.y);
            float rn = fmaf(w21 * Im1, t, p);
            float u  = fmaf(Im1, Im1, c.w);
            u        = fmaf(-rn, rn + i2, u);
            Im1      = fmaf(wo1, u, fmaf(c.x, Im1, -(E * io1)));
            Re1      = rn;
        }
    }

    // tuple output: out[0..n) = Re_f, out[n..2n) = Im_f
    if (v0) { out[e0] = Re0; out[n + e0] = Im0; }
    if (v1) { out[e1] = Re1; out[n + e1] = Im1; }
}

extern "C" void kernel_launch(void* const* d_in, const int* in_sizes, int n_in,
                              void* d_out, int out_size, void* d_ws, size_t ws_size,
                              hipStream_t stream) {
    const float* Re_s  = (const float*)d_in[0];
    const float* Im_s  = (const float*)d_in[1];
    const float* omega = (const float*)d_in[2];
    const float* PiT   = (const float*)d_in[3];
    const float* B     = (const float*)d_in[4];
    float* out = (float*)d_out;

    const int n = in_sizes[0];                     // BATCH
    const int elems_per_block = 2 * TPB;           // 512
    const int blocks = (n + elems_per_block - 1) / elems_per_block;

    metricnet_scan<<<blocks, TPB, 0, stream>>>(Re_s, Im_s, omega, PiT, B, out, n);
}